// GAT_7310034338525
// MI455X (gfx1250) — compile-verified
//
#include <hip/hip_runtime.h>

#define LAYERS 3
#define HC 256   // H*C
#define CD 64    // C == D
#define NEGS 0.2f
#define EPSN 1e-5f

typedef float v2f __attribute__((ext_vector_type(2)));
typedef float v8f __attribute__((ext_vector_type(8)));

__device__ __forceinline__ unsigned f2ord(float f) {
    unsigned u = __float_as_uint(f);
    return (u & 0x80000000u) ? ~u : (u | 0x80000000u);
}
__device__ __forceinline__ float ord2f(unsigned u) {
    return (u & 0x80000000u) ? __uint_as_float(u & 0x7FFFFFFFu) : __uint_as_float(~u);
}
__device__ __forceinline__ float lrelu(float v) { return v > 0.f ? v : NEGS * v; }

__global__ void zero_kernel(float* p, int n) {
    int i = blockIdx.x * blockDim.x + threadIdx.x;
    if (i < n) p[i] = 0.0f;
}

// Y[n,256] = X[n,64] @ W[256,64]^T + bias[256]
// One wave per 16x16 output tile, K=64 via 16 chained f32 WMMAs.
// A-row index is CLAMPED (not predicated) so all loads are unconditional
// global_load_b64 and can be hoisted ahead of the WMMA chain.
__global__ __launch_bounds__(256) void gemm_wmma(
    const float* __restrict__ X, const float* __restrict__ W,
    const float* __restrict__ bias, float* __restrict__ Y, int n) {
    int lane = threadIdx.x;           // 0..31
    int half = lane >> 4;             // 0 or 1
    int l16  = lane & 15;
    int tileM = blockIdx.x;
    int tileO = blockIdx.y * 8 + threadIdx.y;   // 0..15 (256 out channels)
    int row  = tileM * 16 + l16;
    int rclamp = row < n ? row : (n - 1);       // in-bounds; garbage rows never stored
    int ocol = tileO * 16 + l16;
    // A 16x4 f32 layout: lanes0-15 hold K=k0..k0+1; lanes16-31 hold K=k0+2..k0+3
    const float2* xrow = (const float2*)(X + rclamp * CD) + half;  // float2 index step = 2
    // B 4x16 f32: lane = column N; VGPR0 = K rows {k0,k0+2}, VGPR1 = {k0+1,k0+3}
    const float2* wrow = (const float2*)(W + ocol * CD) + half;
    v8f acc = {};
    #pragma unroll
    for (int t = 0; t < 16; t++) {            // k0 = 4*t
        float2 av = xrow[2 * t];
        float2 bv = wrow[2 * t];
        v2f a, b;
        a[0] = av.x; a[1] = av.y;
        b[0] = bv.x; b[1] = bv.y;
        acc = __builtin_amdgcn_wmma_f32_16x16x4_f32(
            false, a, false, b, (short)0, acc, false, false);
    }
    float bb = bias[ocol];
    #pragma unroll
    for (int v = 0; v < 8; v++) {
        int r = tileM * 16 + v + half * 8;    // C/D layout: VGPR v = rows v / v+8
        if (r < n) Y[r * HC + tileO * 16 + l16] = acc[v] + bb;
    }
}

// One wave per edge: logits[e][h] = sum_c att[h][c]*lrelu(xl[src]+xr[dst]);
// also segment-max into m[dst][h] via ordered-uint atomicMax.
__global__ __launch_bounds__(256) void edge_logits(
    const float* __restrict__ xl, const float* __restrict__ xr,
    const float* __restrict__ att, const int* __restrict__ ei,
    int Eorig, int Ep, float* __restrict__ logits, unsigned* __restrict__ m) {
    int w = threadIdx.x >> 5;
    int lane = threadIdx.x & 31;
    int e = blockIdx.x * 8 + w;
    if (e >= Ep) return;
    int src, dst;
    if (e < Eorig) { src = ei[e]; dst = ei[Eorig + e]; }
    else           { src = dst = e - Eorig; }
    int i0 = lane * 8;                       // 8 contiguous (h,c) slots per lane
    const float4* pl = (const float4*)(xl + src * HC + i0);
    const float4* pr = (const float4*)(xr + dst * HC + i0);
    const float4* pa = (const float4*)(att + i0);
    float s = 0.f;
    #pragma unroll
    for (int q = 0; q < 2; q++) {
        float4 a = pl[q], b = pr[q], c = pa[q];
        s += c.x * lrelu(a.x + b.x) + c.y * lrelu(a.y + b.y)
           + c.z * lrelu(a.z + b.z) + c.w * lrelu(a.w + b.w);
    }
    // reduce within each 8-lane head group
    s += __shfl_down(s, 4, 8);
    s += __shfl_down(s, 2, 8);
    s += __shfl_down(s, 1, 8);
    if ((lane & 7) == 0) {
        int h = lane >> 3;
        logits[e * 4 + h] = s;
        atomicMax(m + dst * 4 + h, f2ord(s));
    }
}

// a = exp(logit - max); denom[dst][h] += a  (in-place on logits buffer)
__global__ void edge_exp(float* __restrict__ a, const unsigned* __restrict__ m,
                         const int* __restrict__ ei, int Eorig, int Ep,
                         float* __restrict__ denom) {
    int idx = blockIdx.x * blockDim.x + threadIdx.x;
    if (idx >= Ep * 4) return;
    int e = idx >> 2, h = idx & 3;
    int dst = (e < Eorig) ? ei[Eorig + e] : (e - Eorig);
    float v = expf(a[idx] - ord2f(m[dst * 4 + h]));
    a[idx] = v;
    atomicAdd(denom + dst * 4 + h, v);
}

// agg[dst][h][c] += xl[src][h][c] * a[e][h]/denom[dst][h]
__global__ __launch_bounds__(256) void edge_scatter(
    const float* __restrict__ xl, const float* __restrict__ a,
    const float* __restrict__ denom, const int* __restrict__ ei,
    int Eorig, int Ep, float* __restrict__ agg) {
    int w = threadIdx.x >> 5;
    int lane = threadIdx.x & 31;
    int e = blockIdx.x * 8 + w;
    if (e >= Ep) return;
    int src, dst;
    if (e < Eorig) { src = ei[e]; dst = ei[Eorig + e]; }
    else           { src = dst = e - Eorig; }
    int h = lane >> 3;
    float alpha = a[e * 4 + h] / denom[dst * 4 + h];
    const float* ps = xl + src * HC + lane * 8;
    float* pd = agg + dst * HC + lane * 8;
    #pragma unroll
    for (int j = 0; j < 8; j++) atomicAdd(pd + j, ps[j] * alpha);
}

// h[n][c] = mean over heads of agg + conv_bias; accumulate per-channel sum/sumsq
__global__ __launch_bounds__(256) void head_mean_stats(
    const float* __restrict__ agg, const float* __restrict__ cbias,
    float* __restrict__ h, int n, float* __restrict__ sum, float* __restrict__ sumsq) {
    int c = threadIdx.x & 63;
    int rg = threadIdx.x >> 6;
    float bv = cbias[c];
    float ls = 0.f, lq = 0.f;
    for (int row = blockIdx.x * 4 + rg; row < n; row += gridDim.x * 4) {
        const float* p = agg + row * HC + c;
        float v = 0.25f * (p[0] + p[64] + p[128] + p[192]) + bv;
        h[row * 64 + c] = v;
        ls += v; lq += v * v;
    }
    __shared__ float ss[256], sq[256];
    ss[threadIdx.x] = ls; sq[threadIdx.x] = lq;
    __syncthreads();
    if (threadIdx.x < 64) {
        float ts = ss[threadIdx.x] + ss[threadIdx.x + 64] + ss[threadIdx.x + 128] + ss[threadIdx.x + 192];
        float tq = sq[threadIdx.x] + sq[threadIdx.x + 64] + sq[threadIdx.x + 128] + sq[threadIdx.x + 192];
        atomicAdd(sum + threadIdx.x, ts);
        atomicAdd(sumsq + threadIdx.x, tq);
    }
}

// GraphNorm: var = E[h^2] - (2s - s^2) mu^2 ; stats = {shift = s*mu, scale = rsqrt(var+eps)*w}
__global__ void gn_finalize(const float* __restrict__ sum, const float* __restrict__ sumsq,
                            const float* __restrict__ gs, const float* __restrict__ gw,
                            int n, float* __restrict__ stats) {
    int c = threadIdx.x;   // 64 threads
    float inv_n = 1.0f / (float)n;
    float mu = sum[c] * inv_n;
    float s = gs[c];
    float var = sumsq[c] * inv_n - (2.f * s - s * s) * mu * mu;
    stats[c] = s * mu;
    stats[64 + c] = rsqrtf(var + EPSN) * gw[c];
}

__global__ void gn_apply(const float* __restrict__ h, const float* __restrict__ stats,
                         const float* __restrict__ gb, float* __restrict__ out, int n) {
    int i = blockIdx.x * blockDim.x + threadIdx.x;
    if (i >= n * 64) return;
    int c = i & 63;
    out[i] = (h[i] - stats[c]) * stats[64 + c] + gb[c];
}

extern "C" void kernel_launch(void* const* d_in, const int* in_sizes, int n_in,
                              void* d_out, int out_size, void* d_ws, size_t ws_size,
                              hipStream_t stream) {
    const float* x0  = (const float*)d_in[0];
    const int*   ei  = (const int*)d_in[1];
    const float* Wl  = (const float*)d_in[2];
    const float* bl  = (const float*)d_in[3];
    const float* Wr  = (const float*)d_in[4];
    const float* br  = (const float*)d_in[5];
    const float* att = (const float*)d_in[6];
    const float* cb  = (const float*)d_in[7];
    const float* gw  = (const float*)d_in[8];
    const float* gs  = (const float*)d_in[9];
    const float* gb  = (const float*)d_in[10];

    int n     = in_sizes[0] / CD;
    int Eorig = in_sizes[1] / 2;
    int Ep    = Eorig + n;

    float* ws = (float*)d_ws;
    size_t off = 0;
    float* xl    = ws + off; off += (size_t)n * HC;
    float* xr    = ws + off; off += (size_t)n * HC;
    // contiguous zero-init region: agg | denom | m | sum | sumsq
    float* agg   = ws + off; off += (size_t)n * HC;
    float* denom = ws + off; off += (size_t)n * 4;
    unsigned* mx = (unsigned*)(ws + off); off += (size_t)n * 4;
    float* sum   = ws + off; off += 64;
    float* sumsq = ws + off; off += 64;
    float* logits = ws + off; off += (size_t)Ep * 4;
    float* h     = ws + off; off += (size_t)n * 64;
    float* xcur  = ws + off; off += (size_t)n * 64;
    float* stats = ws + off; off += 128;

    int zcount = n * HC + n * 8 + 128;
    int tilesM = (n + 15) / 16;
    int eblocks = (Ep + 7) / 8;
    const float* xin = x0;

    for (int l = 0; l < LAYERS; l++) {
        zero_kernel<<<(zcount + 255) / 256, 256, 0, stream>>>(agg, zcount);
        dim3 gg(tilesM, 2), gbk(32, 8);
        gemm_wmma<<<gg, gbk, 0, stream>>>(xin, Wl + (size_t)l * HC * CD, bl + l * HC, xl, n);
        gemm_wmma<<<gg, gbk, 0, stream>>>(xin, Wr + (size_t)l * HC * CD, br + l * HC, xr, n);
        edge_logits<<<eblocks, 256, 0, stream>>>(xl, xr, att + l * HC, ei, Eorig, Ep, logits, mx);
        edge_exp<<<(Ep * 4 + 255) / 256, 256, 0, stream>>>(logits, mx, ei, Eorig, Ep, denom);
        edge_scatter<<<eblocks, 256, 0, stream>>>(xl, logits, denom, ei, Eorig, Ep, agg);
        head_mean_stats<<<128, 256, 0, stream>>>(agg, cb + l * 64, h, n, sum, sumsq);
        gn_finalize<<<1, 64, 0, stream>>>(sum, sumsq, gs + l * 64, gw + l * 64, n, stats);
        float* outp = (l == LAYERS - 1) ? (float*)d_out : xcur;
        gn_apply<<<(n * 64 + 255) / 256, 256, 0, stream>>>(h, stats, gb + l * 64, outp, n);
        xin = xcur;
    }
}